// DepthOffset_54082228191691
// MI455X (gfx1250) — compile-verified
//
#include <hip/hip_runtime.h>

// Problem constants (fixed by the reference: depth (4,1,480,640) f32, out (4,18,480,640) i32)
#define BATCH 4
#define H     480
#define W     640
#define TY    32
#define TX    32
#define HALO  6                      // max sample offset: 4 (tap) + 2 (j) = 6
#define LROWS (TY + 2 * HALO)        // 44
#define LCOLS (TX + 2 * HALO)        // 44
#define LSTRIDE 48                   // padded LDS row stride (floats)
#define NTHREADS 256

#if defined(__gfx1250__) && __has_builtin(__builtin_amdgcn_global_load_async_to_lds_b32)
#define HAVE_ASYNC_LDS 1
#else
#define HAVE_ASYNC_LDS 0
#endif

typedef __attribute__((address_space(1))) int GInt;   // global int
typedef __attribute__((address_space(3))) int LInt;   // LDS int

// Per-tile compute: each thread owns 4 consecutive x pixels => b128 channel stores.
// EDGE=false (interior tiles, 234/300): tap in-bounds checks compile away.
template <bool EDGE>
__device__ __forceinline__ void compute_tile(const float* lds, int* __restrict__ outp,
                                             int tid, int x0, int y0)
{
    const int HW  = H * W;
    const int tq  = tid & 7;     // x-quad index within 32-wide tile (8 per row)
    const int ly  = tid >> 3;    // tile row 0..31
    const int lxb = tq << 2;     // base local x (multiple of 4)
    const int y   = y0 + ly;
    const int xb  = x0 + lxb;

    int resH[9][4];
    int resW[9][4];

    #pragma unroll
    for (int p = 0; p < 4; ++p) {
        const int lx = lxb + p;
        const int x  = xb + p;

        const float c    = lds[(ly + HALO) * LSTRIDE + (lx + HALO)];
        const float absc = __builtin_fabsf(c);   // score when tap position OOB (Sp pad => sample 0)

        // 7x7 even-offset score window; (t,j) sample -> sc[2*tr+jr][2*tc+jc].
        // LDS zero-fill reproduces the dp pad. Even/odd p share columns -> compiler CSE.
        float sc[7][7];
        #pragma unroll
        for (int wy = 0; wy < 7; ++wy)
            #pragma unroll
            for (int wx = 0; wx < 7; ++wx)
                sc[wy][wx] = __builtin_fabsf(lds[(ly + 2 * wy) * LSTRIDE + (lx + 2 * wx)] - c);

        #pragma unroll
        for (int t = 0; t < 9; ++t) {
            const int tr = t / 3, tc = t % 3;
            bool ins = true;
            if (EDGE) {
                const int ys = y + 4 * (tr - 1);
                const int xs = x + 4 * (tc - 1);
                ins = ((unsigned)ys < (unsigned)H) && ((unsigned)xs < (unsigned)W);
            }
            float best = 3.402823466e38f;
            int   bj   = 0;
            #pragma unroll
            for (int j = 0; j < 9; ++j) {
                // Edge-direction masks (reference sets these to +inf):
                if ((t == 1 || t == 7) && !(j == 1 || j == 4 || j == 7)) continue;
                if ((t == 3 || t == 5) && !(j == 3 || j == 4 || j == 5)) continue;
                const int jr = j / 3, jc = j % 3;
                const float s = (EDGE && !ins) ? absc : sc[2 * tr + jr][2 * tc + jc];
                if (s < best) { best = s; bj = j; }   // strict '<' => first (lowest j) wins ties
            }
            resH[t][p] = (bj / 3 - 1) * 2;
            resW[t][p] = (bj % 3 - 1) * 2;
        }
    }

    // 18 aligned b128 stores per thread (xb is a multiple of 4).
    #pragma unroll
    for (int t = 0; t < 9; ++t) {
        *(int4*)&outp[t * HW + y * W + xb] =
            make_int4(resH[t][0], resH[t][1], resH[t][2], resH[t][3]);
        *(int4*)&outp[(9 + t) * HW + y * W + xb] =
            make_int4(resW[t][0], resW[t][1], resW[t][2], resW[t][3]);
    }
}

__global__ __launch_bounds__(NTHREADS)
void DepthOffset_kernel(const float* __restrict__ depth, int* __restrict__ out)
{
    __shared__ float lds[LROWS * LSTRIDE];

    const int tid = threadIdx.x;
    const int x0  = blockIdx.x * TX;
    const int y0  = blockIdx.y * TY;
    const int b   = blockIdx.z;

    const float* dplane = depth + (size_t)b * (H * W);

    // ---- Stage depth tile + halo into LDS (CDNA5 async global->LDS, ASYNCcnt) ----
    for (int idx = tid; idx < LROWS * LCOLS; idx += NTHREADS) {
        const int row = idx / LCOLS;
        const int col = idx - row * LCOLS;
        const int gy  = y0 - HALO + row;
        const int gx  = x0 - HALO + col;
        float* lp = &lds[row * LSTRIDE + col];
        if ((unsigned)gy < (unsigned)H && (unsigned)gx < (unsigned)W) {
#if HAVE_ASYNC_LDS
            __builtin_amdgcn_global_load_async_to_lds_b32(
                (GInt*)(dplane + (size_t)gy * W + gx), (LInt*)lp, 0, 0);
#else
            *lp = dplane[(size_t)gy * W + gx];
#endif
        } else {
            *lp = 0.0f;   // zero halo == reference's dp zero padding
        }
    }
#if HAVE_ASYNC_LDS
#if __has_builtin(__builtin_amdgcn_s_wait_asynccnt)
    __builtin_amdgcn_s_wait_asynccnt(0);
#else
    asm volatile("s_wait_asynccnt 0" ::: "memory");
#endif
#endif
    __syncthreads();

    int* outp = out + b * 18 * (H * W);

    const bool edge = (x0 < 4) || (x0 + TX > W - 4) || (y0 < 4) || (y0 + TY > H - 4);
    if (edge) compute_tile<true >(lds, outp, tid, x0, y0);
    else      compute_tile<false>(lds, outp, tid, x0, y0);
}

extern "C" void kernel_launch(void* const* d_in, const int* in_sizes, int n_in,
                              void* d_out, int out_size, void* d_ws, size_t ws_size,
                              hipStream_t stream) {
    const float* depth = (const float*)d_in[0];
    int* out = (int*)d_out;   // reference output dtype is int32

    dim3 grid(W / TX, H / TY, BATCH);   // 20 x 15 x 4, exact tiling
    dim3 block(NTHREADS);
    DepthOffset_kernel<<<grid, block, 0, stream>>>(depth, out);
}